// FocalTverskyLoss_17343077941338
// MI455X (gfx1250) — compile-verified
//
#include <hip/hip_runtime.h>
#include <hip/hip_bf16.h>

#define NB 8
#define NC 19
#define NP (512 * 512)          // pixels per batch image
#define BPB 64                  // blocks per batch
#define TPB 256                 // threads per block
#define ITERS (NP / (4 * TPB * BPB))   // 4
#define EPSV 1e-7f
#define ALPHAV 0.7f
#define BETAV 0.3f

typedef float v2f __attribute__((ext_vector_type(2)));
typedef float v8f __attribute__((ext_vector_type(8)));

// ws layout (floats): S[152] | TP[152] | CNT[152]
#define WS_S 0
#define WS_TP 152
#define WS_CNT 304
#define WS_TOT 456

__global__ void ftl_zero_ws(float* ws) {
    int i = threadIdx.x;
    if (i < WS_TOT) ws[i] = 0.0f;
}

__global__ __launch_bounds__(TPB) void ftl_partial(const float* __restrict__ inp,
                                                   const long long* __restrict__ tgt,
                                                   float* __restrict__ ws) {
    __shared__ float M[TPB * 32];   // 32 KB: per-thread S partials, padded to 32 cols
    __shared__ float tpL[NC];
    __shared__ float cntL[NC];

    const int tid = threadIdx.x;
    const int b   = blockIdx.x >> 6;     // BPB = 64 blocks per batch
    const int blk = blockIdx.x & 63;

    if (tid < NC) { tpL[tid] = 0.0f; cntL[tid] = 0.0f; }
    __syncthreads();

    const float*     inB = inp + (size_t)b * NC * NP;
    const long long* tgB = tgt + (size_t)b * NP;

    float S[NC];
#pragma unroll
    for (int c = 0; c < NC; ++c) S[c] = 0.0f;

    for (int it = 0; it < ITERS; ++it) {
        const int q = (it * BPB + blk) * TPB + tid;   // float4 index within batch

        float v[4 * NC];
#pragma unroll
        for (int c = 0; c < NC; ++c) {
            const float4 t4 =
                reinterpret_cast<const float4*>(inB + (size_t)c * NP)[q];
            v[4 * c + 0] = t4.x; v[4 * c + 1] = t4.y;
            v[4 * c + 2] = t4.z; v[4 * c + 3] = t4.w;
        }
        long long tl[4];
#pragma unroll
        for (int j = 0; j < 4; ++j) tl[j] = tgB[(size_t)4 * q + j];

#pragma unroll
        for (int j = 0; j < 4; ++j) {
            const int t = (int)tl[j];
            float mx = v[j];
#pragma unroll
            for (int c = 1; c < NC; ++c) mx = fmaxf(mx, v[4 * c + j]);
            float den = 0.0f, et = 0.0f;
#pragma unroll
            for (int c = 0; c < NC; ++c) {
                const float e = __expf(v[4 * c + j] - mx);
                v[4 * c + j] = e;
                den += e;
                et = (c == t) ? e : et;
            }
            const float rd = 1.0f / den;
#pragma unroll
            for (int c = 0; c < NC; ++c) S[c] = fmaf(v[4 * c + j], rd, S[c]);
            atomicAdd(&tpL[t], et * rd);
            atomicAdd(&cntL[t], 1.0f);
        }
    }

    __syncthreads();   // LDS atomics visible block-wide

    if (tid < NC) {
        atomicAdd(&ws[WS_TP + b * NC + tid], tpL[tid]);
        atomicAdd(&ws[WS_CNT + b * NC + tid], cntL[tid]);
    }

    // Stage per-thread S partials into LDS matrix M[256][32], zero-padded cols.
#pragma unroll
    for (int c = 0; c < NC; ++c) M[tid * 32 + c] = S[c];
#pragma unroll
    for (int c = NC; c < 32; ++c) M[tid * 32 + c] = 0.0f;

    // Fold 256 rows -> 4 rows.
    for (int s = 128; s >= 4; s >>= 1) {
        __syncthreads();
        const int slots = s * 32;
        for (int idx = tid; idx < slots; idx += TPB) M[idx] += M[idx + slots];
    }
    __syncthreads();

    // Wave 0: two f32 WMMAs with all-ones A reduce the remaining 4x32 block.
    // D[m][n] = sum_k B[k][n]; with n = lane&15 the column sum is bijection-
    // robust regardless of the exact k<->(vgpr,half) mapping.
    if (tid < 32) {
        const int n = tid & 15;
        const int h = tid >> 4;
        v2f a; a.x = 1.0f; a.y = 1.0f;
#pragma unroll
        for (int g = 0; g < 2; ++g) {
            const int cb = g * 16;
            v2f bm;
            bm.x = M[(2 * h + 0) * 32 + cb + n];
            bm.y = M[(2 * h + 1) * 32 + cb + n];
            v8f acc = {};
            acc = __builtin_amdgcn_wmma_f32_16x16x4_f32(
                false, a, false, bm, (short)0, acc, false, false);
            const int col = cb + n;
            if (tid < 16 && col < NC)
                atomicAdd(&ws[WS_S + b * NC + col], acc[0]);
        }
    }
}

__global__ void ftl_finalize(const float* __restrict__ ws, float* __restrict__ out) {
    __shared__ float red[TPB];
    const int tid = threadIdx.x;
    float f = 0.0f;
    if (tid < NB * NC) {
        const float Sv = ws[WS_S + tid];
        const float TP = ws[WS_TP + tid];
        const float CN = ws[WS_CNT + tid];
        const float tv = (TP + EPSV) /
                         (TP + ALPHAV * (CN - TP) + BETAV * (Sv - TP) + EPSV);
        f = 1.0f - tv;   // GAMMA == 1.0
    }
    red[tid] = f;
    __syncthreads();
    for (int s = TPB / 2; s > 0; s >>= 1) {
        if (tid < s) red[tid] += red[tid + s];
        __syncthreads();
    }
    if (tid == 0) out[0] = red[0] * (1.0f / NB);
}

extern "C" void kernel_launch(void* const* d_in, const int* in_sizes, int n_in,
                              void* d_out, int out_size, void* d_ws, size_t ws_size,
                              hipStream_t stream) {
    const float*     inp = (const float*)d_in[0];
    const long long* tgt = (const long long*)d_in[1];   // int64 labels
    float* out = (float*)d_out;
    float* ws  = (float*)d_ws;

    ftl_zero_ws<<<1, 512, 0, stream>>>(ws);
    ftl_partial<<<NB * BPB, TPB, 0, stream>>>(inp, tgt, ws);
    ftl_finalize<<<1, TPB, 0, stream>>>(ws, out);
}